// GraphSAGE_52999896432994
// MI455X (gfx1250) — compile-verified
//
#include <hip/hip_runtime.h>
#include <hip/hip_bf16.h>
#include <math.h>

typedef __attribute__((ext_vector_type(2))) float v2f;
typedef __attribute__((ext_vector_type(8))) float v8f;

#define C_DIM 256
#define N0_DIM 247808
#define N1_DIM 22528
#define NB_DIM 2048
#define E0_DIM 225280
#define E1_DIM 20480

// ---------------------------------------------------------------------------
// Zero a float buffer
// ---------------------------------------------------------------------------
__global__ void zero_kernel(float* __restrict__ p, long n) {
    long i = (long)blockIdx.x * blockDim.x + threadIdx.x;
    if (i < n) p[i] = 0.0f;
}

// ---------------------------------------------------------------------------
// Repack a 256x256 weight matrix into pair-interleaved layout:
//   Wp[(k>>1)*512 + col*2 + (k&1)] = W[k*256 + col]
// so a WMMA B-fragment {W[k][col], W[k+1][col]} becomes one coalesced b64.
// ---------------------------------------------------------------------------
__global__ void repack_kernel(const float* __restrict__ W,
                              float* __restrict__ Wp) {
    int idx = blockIdx.x * blockDim.x + threadIdx.x;   // 0 .. 65535
    int k   = idx >> 8;
    int col = idx & 255;
    Wp[(size_t)(k >> 1) * 512 + col * 2 + (k & 1)] = W[idx];
}

// ---------------------------------------------------------------------------
// Scatter: one wave (32 lanes) per edge.
//   agg[dst[e]][:] += w * x[src[e]][:]   (8 f32 atomics per lane)
//   cnt[dst[e]]    += 1                  (lane 0)
// ---------------------------------------------------------------------------
__global__ void scatter_kernel(const float* __restrict__ x,
                               const int* __restrict__ src,
                               const int* __restrict__ dst,
                               const int* __restrict__ eid,
                               const float* __restrict__ ew,
                               float* __restrict__ agg,
                               float* __restrict__ cnt,
                               int E) {
    int wavesPerBlock = blockDim.x >> 5;
    int e = blockIdx.x * wavesPerBlock + (threadIdx.x >> 5);
    if (e >= E) return;
    int lane = threadIdx.x & 31;
    int s = src[e];
    int d = dst[e];
    float w = ew[eid[e]];
    const float4* xr = (const float4*)(x + (size_t)s * C_DIM);
    float* ar = agg + (size_t)d * C_DIM;
    float4 v0 = xr[lane];        // floats [4*lane .. 4*lane+3]
    float4 v1 = xr[lane + 32];   // floats [128+4*lane .. ]
    int o = 4 * lane;
    atomicAdd(ar + o + 0,   v0.x * w);
    atomicAdd(ar + o + 1,   v0.y * w);
    atomicAdd(ar + o + 2,   v0.z * w);
    atomicAdd(ar + o + 3,   v0.w * w);
    atomicAdd(ar + o + 128, v1.x * w);
    atomicAdd(ar + o + 129, v1.y * w);
    atomicAdd(ar + o + 130, v1.z * w);
    atomicAdd(ar + o + 131, v1.w * w);
    if (lane == 0) atomicAdd(cnt + d, 1.0f);
}

// ---------------------------------------------------------------------------
// cnt -> 1 / max(cnt, 1)
// ---------------------------------------------------------------------------
__global__ void inv_kernel(float* __restrict__ cnt, int n) {
    int i = blockIdx.x * blockDim.x + threadIdx.x;
    if (i < n) cnt[i] = 1.0f / fmaxf(cnt[i], 1.0f);
}

// ---------------------------------------------------------------------------
// K-panel of a GEMM accumulated with V_WMMA_F32_16X16X4_F32.
// Arow points at row base of A (already offset by row*256); scale folds in
// the mean division. acc[j] covers output cols [n0+16j, n0+16j+16).
//
// A 16x4 f32 fragment (2 VGPRs, wave32):
//   lanes 0-15 : a.x = A[m][k0],   a.y = A[m][k0+1]
//   lanes 16-31: a.x = A[m][k0+2], a.y = A[m][k0+3]
// B fragment read from pair-interleaved Wp: one b64 per 16-col tile.
// ---------------------------------------------------------------------------
__device__ __forceinline__ void gemm_panel(const float* __restrict__ Arow,
                                           float scale,
                                           const float* __restrict__ Wp,
                                           int n0, int mrow, int half,
                                           v8f acc[4]) {
    const float* ar = Arow + 2 * half;
    // per-lane base into packed W: pair row (kb>>1) = k0/2 + half, col*2
    const float* wbase = Wp + (size_t)half * 512 + (size_t)(n0 + mrow) * 2;
    for (int k0 = 0; k0 < C_DIM; k0 += 4) {
        v2f a = *(const v2f*)(ar + k0);
        a.x *= scale;
        a.y *= scale;
        const float* wb = wbase + (size_t)(k0 >> 1) * 512;
#pragma unroll
        for (int j = 0; j < 4; ++j) {
            v2f b = *(const v2f*)(wb + 32 * j);   // {W[kb][col], W[kb+1][col]}
            acc[j] = __builtin_amdgcn_wmma_f32_16x16x4_f32(
                false, a, false, b, (short)0, acc[j], false, false);
        }
    }
}

// ---------------------------------------------------------------------------
// Fused GraphConv GEMM: out = (agg * inv) @ Wrel + root @ Wroot + bias [,relu]
// One wave computes a 16x64 output strip (4 WMMA accumulators).
// ---------------------------------------------------------------------------
__global__ void gemm_conv_kernel(const float* __restrict__ agg,
                                 const float* __restrict__ inv,
                                 const float* __restrict__ root,
                                 const float* __restrict__ WrelP,
                                 const float* __restrict__ WrootP,
                                 const float* __restrict__ bias,
                                 float* __restrict__ out,
                                 int M, int do_relu) {
    int wid = (blockIdx.x * blockDim.x + threadIdx.x) >> 5;   // wave-uniform
    int tileM = wid >> 2;          // 16-row tile
    int strip = wid & 3;           // 64-col strip
    if (tileM * 16 >= M) return;   // uniform across the wave: EXEC stays full

    int lane = threadIdx.x & 31;
    int half = lane >> 4;
    int mrow = lane & 15;
    int row = tileM * 16 + mrow;
    int n0 = strip * 64;

    float invRow = inv[row];

    v8f acc[4] = {};
    gemm_panel(agg  + (size_t)row * C_DIM, invRow, WrelP,  n0, mrow, half, acc);
    gemm_panel(root + (size_t)row * C_DIM, 1.0f,   WrootP, n0, mrow, half, acc);

    // C/D 16x16 f32 layout: lanes 0-15 -> M = i (VGPR i), N = lane;
    //                       lanes 16-31 -> M = 8+i,       N = lane-16.
    int orow = tileM * 16 + 8 * half;
#pragma unroll
    for (int j = 0; j < 4; ++j) {
        int col = n0 + 16 * j + mrow;
        float bj = bias[col];
#pragma unroll
        for (int i = 0; i < 8; ++i) {
            float v = acc[j][i] + bj;
            if (do_relu) v = fmaxf(v, 0.0f);
            out[(size_t)(orow + i) * C_DIM + col] = v;
        }
    }
}

// ---------------------------------------------------------------------------
// log_softmax per row of 256; one 256-thread block per row.
// ---------------------------------------------------------------------------
__global__ void logsoftmax_kernel(const float* __restrict__ h,
                                  float* __restrict__ out) {
    __shared__ float red[8];
    int r = blockIdx.x;
    int t = threadIdx.x;
    float v = h[(size_t)r * C_DIM + t];

    // row max
    float m = v;
    for (int o = 16; o > 0; o >>= 1) m = fmaxf(m, __shfl_xor(m, o));
    if ((t & 31) == 0) red[t >> 5] = m;
    __syncthreads();
    float mm = red[0];
#pragma unroll
    for (int i = 1; i < 8; ++i) mm = fmaxf(mm, red[i]);
    __syncthreads();

    // sum of exp
    float s = __expf(v - mm);
    float ss = s;
    for (int o = 16; o > 0; o >>= 1) ss += __shfl_xor(ss, o);
    if ((t & 31) == 0) red[t >> 5] = ss;
    __syncthreads();
    float tot = 0.0f;
#pragma unroll
    for (int i = 0; i < 8; ++i) tot += red[i];

    out[(size_t)r * C_DIM + t] = v - mm - logf(tot);
}

// ---------------------------------------------------------------------------
// Launch
// ---------------------------------------------------------------------------
extern "C" void kernel_launch(void* const* d_in, const int* in_sizes, int n_in,
                              void* d_out, int out_size, void* d_ws, size_t ws_size,
                              hipStream_t stream) {
    const float* x       = (const float*)d_in[0];
    const int*   src0    = (const int*)d_in[1];
    const int*   dst0    = (const int*)d_in[2];
    const int*   eid0    = (const int*)d_in[3];
    const int*   src1    = (const int*)d_in[4];
    const int*   dst1    = (const int*)d_in[5];
    const int*   eid1    = (const int*)d_in[6];
    const float* ew      = (const float*)d_in[7];
    const float* W_rel0  = (const float*)d_in[8];
    const float* b0      = (const float*)d_in[9];
    const float* W_root0 = (const float*)d_in[10];
    const float* W_rel1  = (const float*)d_in[11];
    const float* b1      = (const float*)d_in[12];
    const float* W_root1 = (const float*)d_in[13];

    float* out = (float*)d_out;          // [0 .. NB*C) = h2 ; [NB*C ..) = log_softmax
    float* h2  = out;
    float* lsm = out + (size_t)NB_DIM * C_DIM;

    // workspace layout (floats)
    float* ws    = (float*)d_ws;
    float* agg0  = ws;                                        // N1*C
    float* cnt0  = agg0 + (size_t)N1_DIM * C_DIM;             // N1
    float* agg1  = cnt0 + N1_DIM;                             // NB*C
    float* cnt1  = agg1 + (size_t)NB_DIM * C_DIM;             // NB
    float* h1    = cnt1 + NB_DIM;                             // N1*C
    float* Wp    = h1   + (size_t)N1_DIM * C_DIM;             // 4 * 64K packed weights
    float* Wrel0P  = Wp;
    float* Wroot0P = Wp + 65536;
    float* Wrel1P  = Wp + 2 * 65536;
    float* Wroot1P = Wp + 3 * 65536;
    long zeroN = (long)N1_DIM * C_DIM + N1_DIM + (long)NB_DIM * C_DIM + NB_DIM;

    // 0) pack weights into pair-interleaved layout (one b64 per B-fragment)
    repack_kernel<<<256, 256, 0, stream>>>(W_rel0,  Wrel0P);
    repack_kernel<<<256, 256, 0, stream>>>(W_root0, Wroot0P);
    repack_kernel<<<256, 256, 0, stream>>>(W_rel1,  Wrel1P);
    repack_kernel<<<256, 256, 0, stream>>>(W_root1, Wroot1P);

    // 1) zero aggregate + count buffers (agg0,cnt0,agg1,cnt1 are contiguous)
    zero_kernel<<<(unsigned)((zeroN + 255) / 256), 256, 0, stream>>>(agg0, zeroN);

    // 2) layer-0 scatter: wave per edge
    {
        int wavesPerBlock = 8;
        int blocks = (E0_DIM + wavesPerBlock - 1) / wavesPerBlock;
        scatter_kernel<<<blocks, 256, 0, stream>>>(x, src0, dst0, eid0, ew,
                                                   agg0, cnt0, E0_DIM);
    }

    // 3) inv0
    inv_kernel<<<(N1_DIM + 255) / 256, 256, 0, stream>>>(cnt0, N1_DIM);

    // 4) layer-0 GEMM (+bias, relu) -> h1
    {
        int waves = (N1_DIM / 16) * 4;         // 1408 M-tiles * 4 strips = 5632
        int blocks = waves / 8;                // 8 waves per 256-thread block
        gemm_conv_kernel<<<blocks, 256, 0, stream>>>(agg0, cnt0, x,
                                                     Wrel0P, Wroot0P, b0,
                                                     h1, N1_DIM, 1);
    }

    // 5) layer-1 scatter over h1
    {
        int wavesPerBlock = 8;
        int blocks = (E1_DIM + wavesPerBlock - 1) / wavesPerBlock;
        scatter_kernel<<<blocks, 256, 0, stream>>>(h1, src1, dst1, eid1, ew,
                                                   agg1, cnt1, E1_DIM);
    }

    // 6) inv1
    inv_kernel<<<(NB_DIM + 255) / 256, 256, 0, stream>>>(cnt1, NB_DIM);

    // 7) layer-1 GEMM (+bias, no relu) -> h2 (first half of d_out)
    {
        int waves = (NB_DIM / 16) * 4;         // 128 * 4 = 512
        int blocks = waves / 8;                // 64 blocks
        gemm_conv_kernel<<<blocks, 256, 0, stream>>>(agg1, cnt1, h1,
                                                     Wrel1P, Wroot1P, b1,
                                                     h2, NB_DIM, 0);
    }

    // 8) log_softmax -> second half of d_out
    logsoftmax_kernel<<<NB_DIM, 256, 0, stream>>>(h2, lsm);
}